// GPNStarSourceModule_5789615915682
// MI455X (gfx1250) — compile-verified
//
#include <hip/hip_runtime.h>
#include <hip/hip_bf16.h>

// ---- problem dims (match reference) ----
#define B_      4
#define L_      512
#define S_      100
#define NG_     19
#define GS_     5
#define VOCAB_  6
#define HID_    512
#define HPOOL_  4
#define D_      64
#define ALL_    256
#define FIREH_  32
#define MAXD_   20.0f

#define BL_     (B_ * L_)        // 2048
#define ROWS_   (BL_ * NG_)      // 38912 = 2432 * 16 exactly
#define NW_     (VOCAB_ * HPOOL_) // 24  (rank of the refactored GEMM)

typedef __attribute__((ext_vector_type(2))) float v2f;
typedef __attribute__((ext_vector_type(8))) float v8f;

// ---------------------------------------------------------------------------
// 1) FIRE bias: bias[s] = silu(rel*w1) . w2  with rel = log(c d + 1)/log(c*20 + 1)
// ---------------------------------------------------------------------------
__global__ __launch_bounds__(128)
void fire_bias_kernel(const float* __restrict__ dist, const float* __restrict__ fire_c,
                      const float* __restrict__ w1,   const float* __restrict__ w2,
                      float* __restrict__ bias) {
    int s = threadIdx.x;
    if (s >= S_) return;
    float c   = fmaxf(fire_c[0], 0.0f);
    float rel = dist[s];                       // TIME_SCALE == 1
    float t   = logf(c * rel + 1.0f) / logf(c * MAXD_ + 1.0f);
    float acc = 0.0f;
#pragma unroll
    for (int j = 0; j < FIREH_; ++j) {
        float x = t * w1[j];
        float h = x / (1.0f + expf(-x));       // silu
        acc += h * w2[j];
    }
    bias[s] = acc;
}

// ---------------------------------------------------------------------------
// 2) Fused tail weight: W2[(v*4+h)][col] = value_emb[v, h*64 : h*64+64] . ffn_w[h*64+k, col]
//    24 x 512 outputs, 64-long dots.
// ---------------------------------------------------------------------------
__global__ __launch_bounds__(256)
void w2_kernel(const float* __restrict__ value_emb, const float* __restrict__ ffn_w,
               float* __restrict__ W2) {
    int idx = blockIdx.x * 256 + threadIdx.x;  // 0 .. 24*512-1
    int row = idx >> 9;                        // v*4 + h
    int col = idx & 511;
    int v = row >> 2, h = row & 3;
    const float* ve = value_emb + v * ALL_ + h * D_;
    const float* fw = ffn_w + (h * D_) * HID_ + col;
    float acc = 0.0f;
#pragma unroll 8
    for (int k = 0; k < D_; ++k) acc += ve[k] * fw[k * HID_];
    W2[idx] = acc;
}

// ---------------------------------------------------------------------------
// 3) Softmax-weight matrix A[r][24]: r = bl*19 + n.
//    p[h][g] = softmax_g(attn_w_emb[id_g][h] + bias[n*5+g])
//    A[r][v*4+h] = sum_g p[h][g] * (id_g == v)     (branch-free, no scratch)
// ---------------------------------------------------------------------------
__global__ __launch_bounds__(256)
void aweights_kernel(const int* __restrict__ ids, const float* __restrict__ attn_w_emb,
                     const float* __restrict__ bias, float* __restrict__ A) {
    int r  = blockIdx.x * 256 + threadIdx.x;   // < 38912 (exact grid)
    int bl = r / NG_;
    int n  = r - bl * NG_;
    const int* idp = ids + bl * S_ + n * GS_;

    int   idg[GS_];
    float bg[GS_];
#pragma unroll
    for (int g = 0; g < GS_; ++g) { idg[g] = idp[g]; bg[g] = bias[n * GS_ + g]; }

    float p[HPOOL_][GS_];
#pragma unroll
    for (int h = 0; h < HPOOL_; ++h) {
        float sc[GS_]; float m = -1e30f;
#pragma unroll
        for (int g = 0; g < GS_; ++g) {
            sc[g] = attn_w_emb[idg[g] * HPOOL_ + h] + bg[g];
            m = fmaxf(m, sc[g]);
        }
        float sum = 0.0f;
#pragma unroll
        for (int g = 0; g < GS_; ++g) { sc[g] = expf(sc[g] - m); sum += sc[g]; }
        float inv = 1.0f / sum;
#pragma unroll
        for (int g = 0; g < GS_; ++g) p[h][g] = sc[g] * inv;
    }

    float* ap = A + r * NW_;
#pragma unroll
    for (int v = 0; v < VOCAB_; ++v)
#pragma unroll
        for (int h = 0; h < HPOOL_; ++h) {
            float acc = 0.0f;
#pragma unroll
            for (int g = 0; g < GS_; ++g) acc += (idg[g] == v) ? p[h][g] : 0.0f;
            ap[v * HPOOL_ + h] = acc;
        }
}

// ---------------------------------------------------------------------------
// 4) Rank-24 GEMM via V_WMMA_F32_16X16X4_F32:
//    out_rows(38912 x 512) = A(38912 x 24) @ W2(24 x 512) + ffn_b
//    One 16x16 tile per wave, 8 waves per block. 2432 M-tiles * 32 N-tiles.
//    Accumulator preloaded with ffn_b so the bias add is free.
// ---------------------------------------------------------------------------
__global__ __launch_bounds__(256)
void gemm_wmma_kernel(const float* __restrict__ A, const float* __restrict__ W2,
                      const float* __restrict__ ffn_b, float* __restrict__ out) {
    const int wave = threadIdx.x >> 5;
    const int lane = threadIdx.x & 31;
    const int tile = blockIdx.x * 8 + wave;    // exact: 9728*8 == 2432*32
    const int nt   = tile & 31;                // N tile (512/16 == 32)
    const int mt   = tile >> 5;                // M tile
    const int row0 = mt * 16;
    const int col  = nt * 16 + (lane & 15);
    const int half = lane >> 4;                // 0: lanes 0-15, 1: lanes 16-31

    v8f acc;
    const float b = ffn_b[col];
#pragma unroll
    for (int i = 0; i < 8; ++i) acc[i] = b;

    const int aRow  = row0 + (lane & 15);
    const int kHalf = half * 2;                // A/B frag: half-wave owns K pairs {0,1} / {2,3}

#pragma unroll
    for (int kt = 0; kt < NW_ / 4; ++kt) {     // 6 k-steps
        const int kb = kt * 4 + kHalf;
        v2f a, bb;
        a.x  = A[aRow * NW_ + kb];
        a.y  = A[aRow * NW_ + kb + 1];
        bb.x = W2[kb * HID_ + col];
        bb.y = W2[(kb + 1) * HID_ + col];
        acc = __builtin_amdgcn_wmma_f32_16x16x4_f32(
            /*neg_a=*/false, a, /*neg_b=*/false, bb,
            /*c_mod=*/(short)0, acc, /*reuse_a=*/false, /*reuse_b=*/false);
    }

    // C/D layout: VGPR i -> M = i + half*8, N = lane%16
#pragma unroll
    for (int i = 0; i < 8; ++i) {
        int r  = row0 + half * 8 + i;          // global row (b,l,n) flattened
        int bl = r / NG_;
        int n  = r - bl * NG_;
        out[(bl * 24 + n) * HID_ + col] = acc[i];
    }
}

// ---------------------------------------------------------------------------
// 5) Singles: out[bl][19+j][:] = embed_tab[ids[bl][95+j]][:]   (float4 gather)
// ---------------------------------------------------------------------------
__global__ __launch_bounds__(256)
void singles_kernel(const int* __restrict__ ids, const float* __restrict__ embed_tab,
                    float* __restrict__ out) {
    int idx = blockIdx.x * 256 + threadIdx.x;  // < 2048*5*128 (exact grid)
    int c4  = idx & 127;                       // float4 column
    int t   = idx >> 7;                        // bl*5 + j
    int j   = t % GS_;
    int bl  = t / GS_;
    int id  = ids[bl * S_ + NG_ * GS_ + j];
    const float4* src = (const float4*)embed_tab;
    ((float4*)out)[(bl * 24 + NG_ + j) * (HID_ / 4) + c4] = src[id * (HID_ / 4) + c4];
}

// ---------------------------------------------------------------------------
extern "C" void kernel_launch(void* const* d_in, const int* in_sizes, int n_in,
                              void* d_out, int out_size, void* d_ws, size_t ws_size,
                              hipStream_t stream) {
    const int*   ids   = (const int*)  d_in[0];
    const float* dist  = (const float*)d_in[1];
    const float* attn  = (const float*)d_in[2];
    const float* vemb  = (const float*)d_in[3];
    const float* ffnw  = (const float*)d_in[4];
    const float* ffnb  = (const float*)d_in[5];
    const float* etab  = (const float*)d_in[6];
    const float* firec = (const float*)d_in[7];
    const float* fw1   = (const float*)d_in[8];
    const float* fw2   = (const float*)d_in[9];
    float* out = (float*)d_out;

    // workspace layout (floats): bias[128] | W2[24*512] | A[38912*24]  ~= 3.8 MB
    float* ws   = (float*)d_ws;
    float* bias = ws;
    float* W2   = ws + 128;
    float* A    = ws + 128 + NW_ * HID_;

    fire_bias_kernel<<<1, 128, 0, stream>>>(dist, firec, fw1, fw2, bias);
    w2_kernel<<<(NW_ * HID_) / 256, 256, 0, stream>>>(vemb, ffnw, W2);
    aweights_kernel<<<ROWS_ / 256, 256, 0, stream>>>(ids, attn, bias, A);
    gemm_wmma_kernel<<<(ROWS_ / 16) * (HID_ / 16) / 8, 256, 0, stream>>>(A, W2, ffnb, out);
    singles_kernel<<<(BL_ * GS_ * (HID_ / 4)) / 256, 256, 0, stream>>>(ids, etab, out);
}